// EdgeRadSolver_80461917323968
// MI455X (gfx1250) — compile-verified
//
#include <hip/hip_runtime.h>
#include <math.h>

#define C_LIGHT 0.29979245f
#define N_OBS 8192
#define N_TRACK 2001
#define SEGS 8
#define OBS_PER_BLOCK 32
#define THREADS 256
#define NODES_PER_SEG 250   /* 2000 intervals / 8 segments; 125 panels each */

#if __has_builtin(__builtin_amdgcn_global_load_async_to_lds_b32)
#define HAVE_ASYNC_LDS 1
#else
#define HAVE_ASYNC_LDS 0
#endif

typedef __attribute__((address_space(1))) int gas_int;   // global ("__device__ int*")
typedef __attribute__((address_space(3))) int las_int;   // LDS    ("__shared__ int*")

__device__ __forceinline__ void stage_row(const float* __restrict__ g,
                                          float* __restrict__ l, int n) {
  for (int k = threadIdx.x; k < n; k += THREADS) {
#if HAVE_ASYNC_LDS
    __builtin_amdgcn_global_load_async_to_lds_b32(
        (gas_int*)(g + k), (las_int*)(l + k), 0, 0);
#else
    l[k] = g[k];
#endif
  }
}

extern "C" __global__ void __launch_bounds__(THREADS)
edge_rad_kernel(const float* __restrict__ coords,
                const float* __restrict__ timev,
                const float* __restrict__ rtrk,
                const float* __restrict__ betav,
                const float* __restrict__ omega_p,
                float* __restrict__ out) {
  // rows: 0=time 1=rx 2=ry 3=rz 4=bx 5=by  (48 KB of the 320 KB WGP LDS)
  __shared__ float trk[6][N_TRACK];

  stage_row(timev,             trk[0], N_TRACK);
  stage_row(rtrk,              trk[1], N_TRACK);
  stage_row(rtrk + N_TRACK,    trk[2], N_TRACK);
  stage_row(rtrk + 2*N_TRACK,  trk[3], N_TRACK);
  stage_row(betav,             trk[4], N_TRACK);
  stage_row(betav + N_TRACK,   trk[5], N_TRACK);

#if HAVE_ASYNC_LDS
#if __has_builtin(__builtin_amdgcn_s_wait_asynccnt)
  __builtin_amdgcn_s_wait_asynccnt(0);
#else
  asm volatile("s_wait_asynccnt 0" ::: "memory");
#endif
#endif
  __syncthreads();

  const int tid = threadIdx.x;
  const int seg = tid & (SEGS - 1);
  const int ob  = tid >> 3;
  const int i   = blockIdx.x * OBS_PER_BLOCK + ob;

  __builtin_prefetch(coords + i, 0, 0);

  const float w     = omega_p[0];
  const float winv  = 1.0f / w;
  const float winv2 = winv * winv;
  const float winv3 = winv2 * winv;

  const float cx = coords[i];
  const float cy = coords[N_OBS + i];
  const float cz = coords[2 * N_OBS + i];

  // phase origin: phase[i,0] = time[0] + |coords - r[:,0]| / c
  const float dx0 = cx - trk[1][0];
  const float dy0 = cy - trk[2][0];
  const float dz0 = cz - trk[3][0];
  const float ph0 = trk[0][0] + sqrtf(dx0*dx0 + dy0*dy0 + dz0*dz0) / C_LIGHT;

  const int tBeg = seg * NODES_PER_SEG;          // even -> panel boundary
  const int tEnd = tBeg + NODES_PER_SEG;         // inclusive
  const int gBeg = (seg == 0) ? 0 : (tBeg - 1);  // halo for central gradient
  const int gEnd = (seg == SEGS - 1) ? (N_TRACK - 1) : (tEnd + 1);

  float realx = 0.f, realy = 0.f, imagx = 0.f, imagy = 0.f;

  // rolling geometry (indices g-1, g-2)
  float ph_m1 = 0.f, ph_m2 = 0.f, tm_m1 = 0.f, tm_m2 = 0.f;
  float rn_m1 = 0.f, nx_m1 = 0.f, ny_m1 = 0.f, bx_m1 = 0.f, by_m1 = 0.f;

  // pending Filon panel state: A = last even node, B = last odd node
  float phA = 0.f, i1xA = 0.f, i1yA = 0.f, i2xA = 0.f, i2yA = 0.f, sA = 0.f, cA = 0.f;
  float phB = 0.f, i1xB = 0.f, i1yB = 0.f, i2xB = 0.f, i2yB = 0.f;
  bool haveA = false;

  auto node = [&](float pht, float rnt, float nxt, float nyt,
                  float bxt, float byt, float grad, int t) {
    const float inv1 = 1.0f / (rnt * grad);
    const float i1x  = (bxt - nxt) * inv1;
    const float i1y  = (byt - nyt) * inv1;
    const float inv2 = C_LIGHT / (w * rnt * rnt * grad);
    const float i2x  = nxt * inv2;
    const float i2y  = nyt * inv2;

    if ((t & 1) == 0) {                      // even node: panel endpoint
      float sB_, cB_;
      sincosf(w * pht, &sB_, &cB_);          // one sincos per 2 nodes
      if (haveA) {
        const float x0 = phA, x1 = phB, x2 = pht;
        const float wa = w * x0, wb = w * x2;
        // cos moments J0..J2 over [x0,x2]
        const float j0c = (sB_ - sA) * winv;
        const float j1c = (cB_ + wb * sB_ - cA - wa * sA) * winv2;
        const float j2c = ((wb*wb - 2.0f) * sB_ + 2.0f * wb * cB_
                         - ((wa*wa - 2.0f) * sA + 2.0f * wa * cA)) * winv3;
        // sin moments
        const float j0s = (cA - cB_) * winv;
        const float j1s = (sB_ - wb * cB_ - sA + wa * cA) * winv2;
        const float j2s = ((2.0f - wb*wb) * cB_ + 2.0f * wb * sB_
                         - (2.0f - wa*wa) * cA - 2.0f * wa * sA) * winv3;

        const float d0 = 1.0f / ((x0 - x1) * (x0 - x2));
        const float d1 = 1.0f / ((x1 - x0) * (x1 - x2));
        const float d2 = 1.0f / ((x0 - x2) * (x2 - x1));

        const float w0c = (x1*x2*j0c - x1*j1c - x2*j1c + j2c) * d0;
        const float w1c = (x0*x2*j0c - x0*j1c - x2*j1c + j2c) * d1;
        const float w2c = (x0*(j1c - x1*j0c) + x1*j1c - j2c) * d2;
        const float w0s = (x1*x2*j0s - x1*j1s - x2*j1s + j2s) * d0;
        const float w1s = (x0*x2*j0s - x0*j1s - x2*j1s + j2s) * d1;
        const float w2s = (x0*(j1s - x1*j0s) + x1*j1s - j2s) * d2;

        realx += (w0c*i1xA + w1c*i1xB + w2c*i1x) + (w0s*i2xA + w1s*i2xB + w2s*i2x);
        realy += (w0c*i1yA + w1c*i1yB + w2c*i1y) + (w0s*i2yA + w1s*i2yB + w2s*i2y);
        imagx += (w0s*i1xA + w1s*i1xB + w2s*i1x) - (w0c*i2xA + w1c*i2xB + w2c*i2x);
        imagy += (w0s*i1yA + w1s*i1yB + w2s*i1y) - (w0c*i2yA + w1c*i2yB + w2c*i2y);
      }
      phA = pht; i1xA = i1x; i1yA = i1y; i2xA = i2x; i2yA = i2y;
      sA = sB_; cA = cB_;
      haveA = true;
    } else {                                 // odd node: panel midpoint
      phB = pht; i1xB = i1x; i1yB = i1y; i2xB = i2x; i2yB = i2y;
    }
  };

  for (int g = gBeg; g <= gEnd; ++g) {
    const float tg  = trk[0][g];
    const float ddx = cx - trk[1][g];
    const float ddy = cy - trk[2][g];
    const float ddz = cz - trk[3][g];
    const float rn  = sqrtf(ddx*ddx + ddy*ddy + ddz*ddz);
    const float ph  = (tg + rn / C_LIGHT) - ph0;
    const float nx  = ddx / rn;
    const float ny  = ddy / rn;
    const float bx  = trk[4][g];
    const float by  = trk[5][g];

    // finalize node t = g-1 (gradient needs lookahead phase ph at g)
    const int t = g - 1;
    if (t >= tBeg) {
      float grad;
      if (t == 0) {
        grad = (ph - ph_m1) / (tg - tm_m1);                       // left edge
      } else {
        const float hs = tm_m1 - tm_m2;
        const float hd = tg  - tm_m1;
        grad = (hs*hs*ph + (hd*hd - hs*hs)*ph_m1 - hd*hd*ph_m2)
             / (hs * hd * (hd + hs));                              // interior
      }
      node(ph_m1, rn_m1, nx_m1, ny_m1, bx_m1, by_m1, grad, t);
    }
    // right edge node t = 2000 (one-sided gradient, no lookahead needed)
    if (g == (N_TRACK - 1) && tEnd == (N_TRACK - 1)) {
      const float grad = (ph - ph_m1) / (tg - tm_m1);
      node(ph, rn, nx, ny, bx, by, grad, g);
    }

    ph_m2 = ph_m1; ph_m1 = ph;
    tm_m2 = tm_m1; tm_m1 = tg;
    rn_m1 = rn; nx_m1 = nx; ny_m1 = ny; bx_m1 = bx; by_m1 = by;
  }

  // reduce the 8 segment partials (contiguous 8-lane groups within wave32)
#pragma unroll
  for (int m = 1; m < SEGS; m <<= 1) {
    realx += __shfl_xor(realx, m, 32);
    realy += __shfl_xor(realy, m, 32);
    imagx += __shfl_xor(imagx, m, 32);
    imagy += __shfl_xor(imagy, m, 32);
  }

  if (seg == 0) {
    out[0 * N_OBS + i] = realx;   // real, x
    out[1 * N_OBS + i] = realy;   // real, y
    out[2 * N_OBS + i] = imagx;   // imag, x
    out[3 * N_OBS + i] = imagy;   // imag, y
  }
}

extern "C" void kernel_launch(void* const* d_in, const int* in_sizes, int n_in,
                              void* d_out, int out_size, void* d_ws, size_t ws_size,
                              hipStream_t stream) {
  const float* coords = (const float*)d_in[0];   // [3, 8192]
  const float* timev  = (const float*)d_in[1];   // [2001]
  const float* rtrk   = (const float*)d_in[2];   // [3, 2001]
  const float* betav  = (const float*)d_in[3];   // [3, 2001]
  const float* omega  = (const float*)d_in[4];   // scalar
  float* out = (float*)d_out;                    // [2,2,8192]

  dim3 grid(N_OBS / OBS_PER_BLOCK);              // 256 blocks
  dim3 block(THREADS);                           // 256 threads = 8 wave32
  hipLaunchKernelGGL(edge_rad_kernel, grid, block, 0, stream,
                     coords, timev, rtrk, betav, omega, out);
}